// PPState_88210038326250
// MI455X (gfx1250) — compile-verified
//
#include <hip/hip_runtime.h>

// Problem constants from the reference
#define B_    2
#define N_    4
#define CF_   256
#define T_    8
#define H_    56
#define W_    56
#define CPOS_ 256
#define COUT_ (CF_ + CPOS_)   // 512
#define BNT_  (B_ * N_ * T_)  // 64

typedef __attribute__((ext_vector_type(2))) float v2f;
typedef __attribute__((ext_vector_type(8))) float v8f;

// ---------------------------------------------------------------------------
// Kernel 1: masked average pool over the bbox crop.
// Bandwidth-bound: reads only the crop region (~51 MB total), coalesced along
// W (innermost, contiguous). One (b,n,t) per block.x, 32 channels per block.y
// group; each wave32 handles 4 channels, lanes sweep crop rows, then a
// __shfl_xor tree reduction collapses the wave.
// ---------------------------------------------------------------------------
__global__ void pool_crop_kernel(const float* __restrict__ frames,
                                 const int* __restrict__ bbox,
                                 float* __restrict__ out) {
    const int bnt  = blockIdx.x;          // 0..63  == ((b*N + n)*T + t)
    const int grp  = blockIdx.y;          // 0..7   (channel group of 32)
    const int wave = threadIdx.x >> 5;    // 0..7
    const int lane = threadIdx.x & 31;

    const int b = bnt / (N_ * T_);
    const int n = (bnt / T_) % N_;
    const int t = bnt % T_;

    // bbox row is 16B-aligned: load y0,x0,y1,x1 in one shot
    const int4 box = ((const int4*)bbox)[bnt];
    const int y0 = box.x, x0 = box.y, y1 = box.z, x1 = box.w;
    const int chh = max(y1 - y0, 0);
    const int cww = max(x1 - x0, 0);
    const int npix = chh * cww;
    const float inv = 1.0f / (float)max(npix, 1);  // degen/empty -> sum 0 / 1 = 0

    #pragma unroll
    for (int i = 0; i < 4; ++i) {
        const int c = grp * 32 + wave * 4 + i;
        const float* base =
            frames + ((((size_t)b * N_ + n) * CF_ + c) * T_ + t) * (size_t)(H_ * W_);
        float s = 0.0f;
        for (int dy = 0; dy < chh; ++dy) {
            const float* rowp = base + (size_t)(y0 + dy) * W_ + x0;
            for (int dx = lane; dx < cww; dx += 32) {
                s += rowp[dx];
            }
        }
        // wave32 tree reduction
        #pragma unroll
        for (int off = 16; off > 0; off >>= 1) {
            s += __shfl_xor(s, off, 32);
        }
        if (lane == 0) {
            out[(size_t)bnt * COUT_ + c] = s * inv;
        }
    }
}

// ---------------------------------------------------------------------------
// Kernel 2: positional embedding  pos = bbox_f32 @ W_pos^T  ([64,4]@[4,256]).
// K=4 is the native shape of V_WMMA_F32_16X16X4_F32. One wave per 16x16 tile:
// 4 M-tiles (rows = flattened (b,n,t)) x 16 N-tiles (cols = C_POS).
//
// 32-bit A 16x4 layout (ISA 7.12.2): lanes 0-15 -> M=lane, VGPR0=K0,VGPR1=K1;
// lanes 16-31 -> M=lane-16, VGPR0=K2,VGPR1=K3. B 4x16 mirrors with N across
// lanes. C/D: VGPR v -> M=v (lanes 0-15) / M=v+8 (lanes 16-31), N=lane&15.
// ---------------------------------------------------------------------------
__global__ void pos_wmma_kernel(const int* __restrict__ bbox,
                                const float* __restrict__ Wpos,
                                float* __restrict__ out) {
    const int tileM = blockIdx.x & 3;     // 4 row tiles  (64 rows)
    const int tileN = blockIdx.x >> 2;    // 16 col tiles (256 cols)
    const int lane  = threadIdx.x;        // 0..31, EXEC all-ones (no divergence)
    const int half  = lane >> 4;          // 0: K=0,1   1: K=2,3
    const int l     = lane & 15;
    const int k0    = half * 2;

    // A fragment: A[m,k] = (float)bbox[row, k]
    const int row = tileM * 16 + l;
    v2f a;
    a.x = (float)bbox[row * 4 + k0];
    a.y = (float)bbox[row * 4 + k0 + 1];

    // B fragment: B[k,n] = W_pos[col, k]  (out = bbox @ W_pos^T)
    const int col = tileN * 16 + l;
    v2f bm;
    bm.x = Wpos[col * 4 + k0];
    bm.y = Wpos[col * 4 + k0 + 1];

    v8f c = {};
    c = __builtin_amdgcn_wmma_f32_16x16x4_f32(
        /*neg_a=*/false, a, /*neg_b=*/false, bm,
        /*c_mod=*/(short)0, c, /*reuse_a=*/false, /*reuse_b=*/false);

    #pragma unroll
    for (int v = 0; v < 8; ++v) {
        const int m = tileM * 16 + v + half * 8;
        out[(size_t)m * COUT_ + CF_ + col] = c[v];
    }
}

// ---------------------------------------------------------------------------
extern "C" void kernel_launch(void* const* d_in, const int* in_sizes, int n_in,
                              void* d_out, int out_size, void* d_ws, size_t ws_size,
                              hipStream_t stream) {
    (void)in_sizes; (void)n_in; (void)out_size; (void)d_ws; (void)ws_size;

    const float* frames = (const float*)d_in[0];  // [B,N,CF,T,H,W] f32
    const int*   bbox   = (const int*)d_in[1];    // [B,N,T,4] int32
    const float* Wpos   = (const float*)d_in[2];  // [C_POS,4] f32
    float*       out    = (float*)d_out;          // [B,N,T,COUT] f32

    // Crop-average pooling: 64 x 8 blocks of 256 threads (8 waves each)
    dim3 gridPool(BNT_, CF_ / 32);
    pool_crop_kernel<<<gridPool, 256, 0, stream>>>(frames, bbox, out);

    // Positional GEMM via f32 WMMA: 64 tiles, one wave32 per tile
    pos_wmma_kernel<<<dim3(64), dim3(32), 0, stream>>>(bbox, Wpos, out);
}